// SEDR_515396075923
// MI455X (gfx1250) — compile-verified
//
#include <hip/hip_runtime.h>
#include <hip/hip_bf16.h>

// ---------------- problem constants (match reference) ----------------
static constexpr int N_SPOTS = 50000;
static constexpr int D_IN    = 3000;
static constexpr int FH1     = 100;
static constexpr int FH2     = 20;
static constexpr int GH1     = 32;
static constexpr int GH2     = 8;
static constexpr int K_CL    = 10;
static constexpr int LAT     = FH2 + GH2;   // 28
static constexpr int E_EDGES = 800000;
static constexpr float BN_EPS = 1e-4f;

typedef __attribute__((ext_vector_type(2))) float v2f;
typedef __attribute__((ext_vector_type(8))) float v8f;

__device__ __forceinline__ float elu1(float v) {
    return v > 0.0f ? v : expm1f(v);
}

// ---------------------------------------------------------------------
// fill zeros (scratch + atomic accumulators must be re-zeroed each call)
// ---------------------------------------------------------------------
__global__ void fill_zero_kernel(float* __restrict__ p, int n) {
    int i = blockIdx.x * 256 + threadIdx.x;
    if (i < n) p[i] = 0.0f;
}

// ---------------------------------------------------------------------
// enc1: feat1 = ELU(BN(x @ W1 + b1))   [N,3000]x[3000,100] -> [N,100]
// WMMA f32 16x16x4. Block: 7 waves (224 thr) x 16-col tiles, 64 rows
// (4 M sub-tiles per wave share each B fragment -> 4x B-load reuse).
// x staged via LDS (64 x 200 chunk, stride padded to 204).
// ---------------------------------------------------------------------
static constexpr int ENC1_KC   = 200;   // K chunk per LDS stage (3000/200 = 15)
static constexpr int ENC1_LDW  = 204;   // padded LDS row stride (conflict-free)
static constexpr int ENC1_ROWS = 64;    // rows per block (4 tiles of 16)

__global__ __launch_bounds__(224) void enc1_kernel(
    const float* __restrict__ x,  const float* __restrict__ W1,
    const float* __restrict__ b1, const float* __restrict__ g1,
    const float* __restrict__ bb1, float* __restrict__ feat1)
{
    __shared__ float sx[ENC1_ROWS * ENC1_LDW];   // ~52 KB

    const int tid  = threadIdx.x;
    const int wave = tid >> 5;          // 0..6  -> column tile
    const int lane = tid & 31;
    const int m0   = blockIdx.x * ENC1_ROWS;
    const int n0   = wave * 16;
    const int colL = n0 + (lane & 15);                 // logical column 0..111
    const int colC = colL < FH1 ? colL : FH1 - 1;      // clamped (safe loads)

    // fused BatchNorm(eval) constants for this lane's column
    const float s1   = g1[colC] * rsqrtf(1.0f + BN_EPS);
    const float bnb  = bb1[colC];
    const float bias = b1[colC];

    const int mrow = lane & 15;
    const int koff = (lane >> 4) * 2;   // A frag: K=0/1 (lanes 0-15), K=2/3 (16-31)

    v8f acc[4];
    #pragma unroll
    for (int rt = 0; rt < 4; ++rt) acc[rt] = (v8f){};

    for (int kb = 0; kb < D_IN; kb += ENC1_KC) {
        __syncthreads();   // previous stage's compute done before overwrite
        // cooperative load x[m0..m0+63, kb..kb+199] as float4 (zero-pad OOB rows)
        for (int i = tid; i < ENC1_ROWS * (ENC1_KC / 4); i += 224) {
            const int r  = i / (ENC1_KC / 4);
            const int c4 = i % (ENC1_KC / 4);
            float4 v = make_float4(0.f, 0.f, 0.f, 0.f);
            if (m0 + r < N_SPOTS)
                v = *(const float4*)(x + (size_t)(m0 + r) * D_IN + kb + c4 * 4);
            float* dst = &sx[r * ENC1_LDW + c4 * 4];
            dst[0] = v.x; dst[1] = v.y; dst[2] = v.z; dst[3] = v.w;
        }
        __syncthreads();

        #pragma unroll 2
        for (int kk = 0; kk < ENC1_KC; kk += 4) {
            // B fragment (4x16 f32): two consecutive K rows of W1 (L2-hot),
            // shared by all 4 M sub-tiles.
            const int krow = kb + kk + koff;
            v2f b;
            b.x = W1[(size_t)krow * FH1 + colC];
            b.y = W1[(size_t)(krow + 1) * FH1 + colC];
            #pragma unroll
            for (int rt = 0; rt < 4; ++rt) {
                const float* ap = &sx[(rt * 16 + mrow) * ENC1_LDW + kk + koff];
                v2f a; a.x = ap[0]; a.y = ap[1];
                acc[rt] = __builtin_amdgcn_wmma_f32_16x16x4_f32(
                    false, a, false, b, (short)0, acc[rt], false, false);
            }
        }
    }

    if (colL < FH1) {
        #pragma unroll
        for (int rt = 0; rt < 4; ++rt) {
            const int rbase = m0 + rt * 16 + (lane >> 4) * 8;  // M = r + 8*(lane>=16)
            #pragma unroll
            for (int r = 0; r < 8; ++r) {
                const int row = rbase + r;
                if (row < N_SPOTS) {
                    float v = (acc[rt][r] + bias) * s1 + bnb;
                    feat1[(size_t)row * FH1 + colL] = elu1(v);
                }
            }
        }
    }
}

// ---------------------------------------------------------------------
// enc2: feat_x = ELU(BN(feat1 @ W2 + b2))  [N,100]x[100,20] -> [N,20]
// writes both feat_x output region and z[:, 0:20]
// ---------------------------------------------------------------------
__global__ __launch_bounds__(256) void enc2_kernel(
    const float* __restrict__ feat1, const float* __restrict__ W2,
    const float* __restrict__ b2, const float* __restrict__ g2,
    const float* __restrict__ bb2,
    float* __restrict__ out_fx, float* __restrict__ out_z)
{
    __shared__ float s[64 * FH1];
    const int tid = threadIdx.x;
    const int r0  = blockIdx.x * 64;
    for (int i = tid; i < 64 * FH1; i += 256) {
        const int rr = r0 + i / FH1;
        s[i] = (rr < N_SPOTS) ? feat1[(size_t)r0 * FH1 + i] : 0.0f;
    }
    __syncthreads();
    const float rs = rsqrtf(1.0f + BN_EPS);
    #pragma unroll
    for (int t = 0; t < 5; ++t) {
        const int idx = t * 256 + tid;          // 0..1279
        const int r = idx / FH2, c = idx % FH2;
        const int row = r0 + r;
        if (row < N_SPOTS) {
            float a = b2[c];
            for (int k = 0; k < FH1; ++k) a += s[r * FH1 + k] * W2[k * FH2 + c];
            const float v = elu1(a * (g2[c] * rs) + bb2[c]);
            out_fx[(size_t)row * FH2 + c] = v;
            out_z [(size_t)row * LAT + c] = v;
        }
    }
}

// ---------------------------------------------------------------------
// gc1 linear: t1 = feat_x @ gc1_W   [N,20]x[20,32] -> [N,32]
// ---------------------------------------------------------------------
__global__ __launch_bounds__(256) void gc1_kernel(
    const float* __restrict__ fx, const float* __restrict__ W,
    float* __restrict__ t1)
{
    __shared__ float s[64 * FH2];
    __shared__ float w[FH2 * GH1];
    const int tid = threadIdx.x;
    const int r0  = blockIdx.x * 64;
    for (int i = tid; i < FH2 * GH1; i += 256) w[i] = W[i];
    for (int i = tid; i < 64 * FH2; i += 256) {
        const int rr = r0 + i / FH2;
        s[i] = (rr < N_SPOTS) ? fx[(size_t)r0 * FH2 + i] : 0.0f;
    }
    __syncthreads();
    #pragma unroll
    for (int t = 0; t < 8; ++t) {
        const int idx = t * 256 + tid;          // 0..2047
        const int r = idx / GH1, c = idx % GH1;
        const int row = r0 + r;
        if (row < N_SPOTS) {
            float a = 0.0f;
            #pragma unroll
            for (int k = 0; k < FH2; ++k) a += s[r * FH2 + k] * w[k * GH1 + c];
            t1[(size_t)row * GH1 + c] = a;
        }
    }
}

// ---------------------------------------------------------------------
// SpMM (COO segment-sum), 32 features: one lane per feature, one edge
// per 32 lanes; atomic scatter-add into pre-zeroed accumulator.
// ---------------------------------------------------------------------
__global__ __launch_bounds__(256) void spmm32_kernel(
    const float* __restrict__ vals, const int* __restrict__ erow,
    const int* __restrict__ ecol, const float* __restrict__ dense,
    float* __restrict__ acc)
{
    const int t = blockIdx.x * 256 + threadIdx.x;
    const int e = t >> 5, f = t & 31;
    if (e < E_EDGES) {
        const int r = erow[e], c = ecol[e];
        const float v = vals[e];
        atomicAdd(&acc[(size_t)r * GH1 + f], v * dense[(size_t)c * GH1 + f]);
    }
}

// ---------------------------------------------------------------------
// h1 = relu(h1acc); t23 = h1 @ [gc2_W | gc3_W]   -> [N,16]
// ---------------------------------------------------------------------
__global__ __launch_bounds__(256) void gc23_kernel(
    const float* __restrict__ h1acc, const float* __restrict__ W2g,
    const float* __restrict__ W3g, float* __restrict__ t23)
{
    __shared__ float w[2 * GH1 * GH2];
    const int tid = threadIdx.x;
    for (int i = tid; i < GH1 * GH2; i += 256) {
        w[i] = W2g[i];
        w[GH1 * GH2 + i] = W3g[i];
    }
    __syncthreads();
    const int idx = blockIdx.x * 256 + tid;
    if (idx < N_SPOTS * 16) {
        const int n = idx >> 4, j = idx & 15;
        const float* wc = (j < 8) ? &w[j] : &w[GH1 * GH2 + (j - 8)];
        float a = 0.0f;
        #pragma unroll
        for (int k = 0; k < GH1; ++k) {
            float h = h1acc[(size_t)n * GH1 + k];
            h = h > 0.0f ? h : 0.0f;
            a += h * wc[k * GH2];
        }
        t23[(size_t)n * 16 + j] = a;
    }
}

// ---------------------------------------------------------------------
// SpMM for mu (features 0-7) and logvar (features 8-15) together.
// ---------------------------------------------------------------------
__global__ __launch_bounds__(256) void spmm16_kernel(
    const float* __restrict__ vals, const int* __restrict__ erow,
    const int* __restrict__ ecol, const float* __restrict__ t23,
    float* __restrict__ mu, float* __restrict__ logvar)
{
    const int t = blockIdx.x * 256 + threadIdx.x;
    const int e = t >> 4, f = t & 15;
    if (e < E_EDGES) {
        const int r = erow[e], c = ecol[e];
        const float d = vals[e] * t23[(size_t)c * 16 + f];
        if (f < 8) atomicAdd(&mu[(size_t)r * GH2 + f], d);
        else       atomicAdd(&logvar[(size_t)r * GH2 + (f - 8)], d);
    }
}

// ---------------------------------------------------------------------
// gnn_z = mu (copy);  z[:, 20:28] = mu
// ---------------------------------------------------------------------
__global__ void finz_kernel(const float* __restrict__ mu,
                            float* __restrict__ gnnz, float* __restrict__ z)
{
    const int i = blockIdx.x * 256 + threadIdx.x;
    if (i < N_SPOTS * GH2) {
        const float v = mu[i];
        gnnz[i] = v;
        const int n = i >> 3, j = i & 7;
        z[(size_t)n * LAT + FH2 + j] = v;
    }
}

// ---------------------------------------------------------------------
// decoder: de_feat = ELU(BN(z @ dec_W + dec_b))  [N,28]x[28,3000]
// WMMA f32 16x16x4: K = 28 = 7 steps. Block = 8 waves, 16 rows; z tile
// in LDS; each wave strides over the 188 column tiles (last masked).
// Store-bandwidth bound (600 MB de_feat).
// ---------------------------------------------------------------------
__global__ __launch_bounds__(256) void dec_kernel(
    const float* __restrict__ z, const float* __restrict__ W,
    const float* __restrict__ b, const float* __restrict__ g,
    const float* __restrict__ bb, float* __restrict__ out)
{
    __shared__ float sz[16 * LAT];
    const int tid = threadIdx.x, wave = tid >> 5, lane = tid & 31;
    const int m0 = blockIdx.x * 16;      // 3125 blocks, exact
    for (int i = tid; i < 16 * LAT; i += 256)
        sz[i] = z[(size_t)m0 * LAT + i];
    __syncthreads();

    const int mrow = lane & 15;
    const int koff = (lane >> 4) * 2;
    v2f afr[7];
    #pragma unroll
    for (int kk = 0; kk < 7; ++kk) {
        const float* ap = &sz[mrow * LAT + kk * 4 + koff];
        afr[kk].x = ap[0]; afr[kk].y = ap[1];
    }
    const float rs = rsqrtf(1.0f + BN_EPS);
    const int NCT = (D_IN + 15) / 16;    // 188

    for (int ct = wave; ct < NCT; ct += 8) {
        const int n0   = ct * 16;
        const int colL = n0 + (lane & 15);
        const int colC = colL < D_IN ? colL : D_IN - 1;
        v8f acc = {};
        #pragma unroll
        for (int kk = 0; kk < 7; ++kk) {
            const int krow = kk * 4 + koff;
            v2f bfr;
            bfr.x = W[(size_t)krow * D_IN + colC];
            bfr.y = W[(size_t)(krow + 1) * D_IN + colC];
            acc = __builtin_amdgcn_wmma_f32_16x16x4_f32(
                false, afr[kk], false, bfr, (short)0, acc, false, false);
        }
        if (colL < D_IN) {
            const float s3 = g[colC] * rs, bnb = bb[colC], bv = b[colC];
            const int rbase = m0 + (lane >> 4) * 8;
            #pragma unroll
            for (int r = 0; r < 8; ++r) {
                const float v = (acc[r] + bv) * s3 + bnb;
                out[(size_t)(rbase + r) * D_IN + colL] = elu1(v);
            }
        }
    }
}

// ---------------------------------------------------------------------
// DEC soft assignment q (Student-t, alpha=1)
// ---------------------------------------------------------------------
__global__ __launch_bounds__(256) void q_kernel(
    const float* __restrict__ z, const float* __restrict__ cluster,
    float* __restrict__ q)
{
    __shared__ float cl[K_CL * LAT];
    const int tid = threadIdx.x;
    for (int i = tid; i < K_CL * LAT; i += 256) cl[i] = cluster[i];
    __syncthreads();
    const int n = blockIdx.x * 256 + tid;
    if (n < N_SPOTS) {
        float zr[LAT];
        #pragma unroll
        for (int j = 0; j < LAT; ++j) zr[j] = z[(size_t)n * LAT + j];
        float qs[K_CL], sum = 0.0f;
        #pragma unroll
        for (int k = 0; k < K_CL; ++k) {
            float d = 0.0f;
            #pragma unroll
            for (int j = 0; j < LAT; ++j) {
                const float t = zr[j] - cl[k * LAT + j];
                d += t * t;
            }
            const float qq = 1.0f / (1.0f + d);
            qs[k] = qq; sum += qq;
        }
        const float inv = 1.0f / sum;
        #pragma unroll
        for (int k = 0; k < K_CL; ++k) q[(size_t)n * K_CL + k] = qs[k] * inv;
    }
}

// ---------------------------------------------------------------------
extern "C" void kernel_launch(void* const* d_in, const int* in_sizes, int n_in,
                              void* d_out, int out_size, void* d_ws, size_t ws_size,
                              hipStream_t stream)
{
    (void)in_sizes; (void)n_in; (void)out_size; (void)ws_size;

    const float* x     = (const float*)d_in[0];
    const float* adj   = (const float*)d_in[1];
    const float* W1    = (const float*)d_in[2];
    const float* b1    = (const float*)d_in[3];
    const float* g1    = (const float*)d_in[4];
    const float* bb1   = (const float*)d_in[5];
    const float* W2    = (const float*)d_in[6];
    const float* b2    = (const float*)d_in[7];
    const float* g2    = (const float*)d_in[8];
    const float* bb2   = (const float*)d_in[9];
    const float* gc1W  = (const float*)d_in[10];
    const float* gc2W  = (const float*)d_in[11];
    const float* gc3W  = (const float*)d_in[12];
    const float* decW  = (const float*)d_in[13];
    const float* decb  = (const float*)d_in[14];
    const float* g3    = (const float*)d_in[15];
    const float* bb3   = (const float*)d_in[16];
    const float* clus  = (const float*)d_in[17];
    const int*   erow  = (const int*)d_in[18];
    const int*   ecol  = (const int*)d_in[19];

    float* out   = (float*)d_out;
    float* o_z   = out;                               // [N,28]
    float* o_mu  = o_z  + (size_t)N_SPOTS * LAT;      // [N,8]
    float* o_lv  = o_mu + (size_t)N_SPOTS * GH2;      // [N,8]
    float* o_de  = o_lv + (size_t)N_SPOTS * GH2;      // [N,3000]
    float* o_q   = o_de + (size_t)N_SPOTS * D_IN;     // [N,10]
    float* o_fx  = o_q  + (size_t)N_SPOTS * K_CL;     // [N,20]
    float* o_gz  = o_fx + (size_t)N_SPOTS * FH2;      // [N,8]

    float* ws    = (float*)d_ws;
    float* feat1 = ws;                                // [N,100]
    float* t1    = feat1 + (size_t)N_SPOTS * FH1;     // [N,32]
    float* h1acc = t1    + (size_t)N_SPOTS * GH1;     // [N,32]
    float* t23   = h1acc + (size_t)N_SPOTS * GH1;     // [N,16]

    // re-zero atomic accumulators every call (graph-replay safe)
    fill_zero_kernel<<<(N_SPOTS * GH1 + 255) / 256, 256, 0, stream>>>(h1acc, N_SPOTS * GH1);
    fill_zero_kernel<<<(N_SPOTS * GH2 + 255) / 256, 256, 0, stream>>>(o_mu, N_SPOTS * GH2);
    fill_zero_kernel<<<(N_SPOTS * GH2 + 255) / 256, 256, 0, stream>>>(o_lv, N_SPOTS * GH2);

    enc1_kernel<<<(N_SPOTS + ENC1_ROWS - 1) / ENC1_ROWS, 224, 0, stream>>>(
        x, W1, b1, g1, bb1, feat1);
    enc2_kernel<<<(N_SPOTS + 63) / 64, 256, 0, stream>>>(feat1, W2, b2, g2, bb2, o_fx, o_z);
    gc1_kernel<<<(N_SPOTS + 63) / 64, 256, 0, stream>>>(o_fx, gc1W, t1);
    spmm32_kernel<<<(E_EDGES * 32) / 256, 256, 0, stream>>>(adj, erow, ecol, t1, h1acc);
    gc23_kernel<<<(N_SPOTS * 16 + 255) / 256, 256, 0, stream>>>(h1acc, gc2W, gc3W, t23);
    spmm16_kernel<<<(E_EDGES * 16) / 256, 256, 0, stream>>>(adj, erow, ecol, t23, o_mu, o_lv);
    finz_kernel<<<(N_SPOTS * GH2 + 255) / 256, 256, 0, stream>>>(o_mu, o_gz, o_z);
    dec_kernel<<<N_SPOTS / 16, 256, 0, stream>>>(o_z, decW, decb, g3, bb3, o_de);
    q_kernel<<<(N_SPOTS + 255) / 256, 256, 0, stream>>>(o_z, clus, o_q);
}